// joint_model_61332132987122
// MI455X (gfx1250) — compile-verified
//
#include <hip/hip_runtime.h>

// Problem constants (from reference)
#define NXc  50000      // xenium cells
#define NVc  4000       // visium spots
#define NFc  50         // factors
#define NFP  64         // factors padded to WMMA tile multiple
#define NGc  18000      // genes
#define NGXc 300        // xenium panel genes
#define NEc  2000000LL  // edges
#define KSPLIT 50       // K-chunks for gamma^T GEMM (1000 rows each)

typedef float v2f __attribute__((ext_vector_type(2)));
typedef float v8f __attribute__((ext_vector_type(8)));

// ---------- order-preserving float <-> uint encoding for atomic max ----------
__device__ __forceinline__ unsigned encf(float f) {
  unsigned u = __float_as_uint(f);
  return (u & 0x80000000u) ? ~u : (u | 0x80000000u);
}
__device__ __forceinline__ float decf(unsigned u) {
  return (u & 0x80000000u) ? __uint_as_float(u & 0x7fffffffu)
                           : __uint_as_float(~u);
}

// ---------------------------- init workspace --------------------------------
__global__ void k_init(float* gacc, float* denom, unsigned* menc, float* vis) {
  int i = blockIdx.x * blockDim.x + threadIdx.x;
  int stride = gridDim.x * blockDim.x;
  for (int j = i; j < NXc * NFP; j += stride) gacc[j] = 0.0f;
  for (int j = i; j < NVc * NFP; j += stride) vis[j] = 0.0f;   // split-K accum target
  for (int j = i; j < NXc; j += stride) {
    denom[j] = 0.0f;
    menc[j] = encf(-3.0e38f);
  }
}

// ------------- per-node: softmax(Px) -> h = s @ linW^T, att dots ------------
__global__ void k_node(const float* __restrict__ Px, const float* __restrict__ linW,
                       const float* __restrict__ attS, const float* __restrict__ attD,
                       float* __restrict__ h, float* __restrict__ asrc,
                       float* __restrict__ adst) {
  __shared__ float sW[NFc * NFc];
  __shared__ float sAS[NFc], sAD[NFc];
  for (int j = threadIdx.x; j < NFc * NFc; j += blockDim.x) sW[j] = linW[j];
  if (threadIdx.x < NFc) {
    sAS[threadIdx.x] = attS[threadIdx.x];
    sAD[threadIdx.x] = attD[threadIdx.x];
  }
  __syncthreads();
  int n = blockIdx.x * blockDim.x + threadIdx.x;
  if (n >= NXc) return;
  float s[NFc];
  float mx = -3.0e38f;
#pragma unroll
  for (int j = 0; j < NFc; ++j) { s[j] = Px[n * NFc + j]; mx = fmaxf(mx, s[j]); }
  float sum = 0.0f;
#pragma unroll
  for (int j = 0; j < NFc; ++j) { s[j] = __expf(s[j] - mx); sum += s[j]; }
  float inv = 1.0f / sum;
#pragma unroll
  for (int j = 0; j < NFc; ++j) s[j] *= inv;
  float as = 0.0f, ad = 0.0f;
#pragma unroll 2
  for (int i = 0; i < NFc; ++i) {
    float acc = 0.0f;
#pragma unroll
    for (int j = 0; j < NFc; ++j) acc += s[j] * sW[i * NFc + j];
    h[(size_t)n * NFP + i] = acc;
    as += acc * sAS[i];
    ad += acc * sAD[i];
  }
  asrc[n] = as;
  adst[n] = ad;
}

// ------------------------------ edge helpers --------------------------------
__device__ __forceinline__ void edge_pair(const long long* __restrict__ ei,
                                          long long e, int& s, int& d) {
  if (e < NEc) { s = (int)ei[e]; d = (int)ei[NEc + e]; }
  else         { s = d = (int)(e - NEc); }  // self loops
}
__device__ __forceinline__ float edge_logit(const float* asrc, const float* adst,
                                            int s, int d) {
  float x = asrc[s] + adst[d];
  return x > 0.0f ? x : 0.2f * x;  // leaky_relu 0.2
}

__global__ void k_edge_max(const long long* __restrict__ ei,
                           const float* __restrict__ asrc,
                           const float* __restrict__ adst,
                           unsigned* __restrict__ menc) {
  long long i = (long long)blockIdx.x * blockDim.x + threadIdx.x;
  long long stride = (long long)gridDim.x * blockDim.x;
  for (long long e = i; e < NEc + NXc; e += stride) {
    int s, d; edge_pair(ei, e, s, d);
    atomicMax(&menc[d], encf(edge_logit(asrc, adst, s, d)));
  }
}

__global__ void k_edge_den(const long long* __restrict__ ei,
                           const float* __restrict__ asrc,
                           const float* __restrict__ adst,
                           const unsigned* __restrict__ menc,
                           float* __restrict__ denom) {
  long long i = (long long)blockIdx.x * blockDim.x + threadIdx.x;
  long long stride = (long long)gridDim.x * blockDim.x;
  for (long long e = i; e < NEc + NXc; e += stride) {
    int s, d; edge_pair(ei, e, s, d);
    float l = edge_logit(asrc, adst, s, d);
    atomicAdd(&denom[d], __expf(l - decf(menc[d])));
  }
}

// one WAVE per edge: lanes cover the 50 factor slots (coalesced atomics)
__global__ void k_edge_acc(const long long* __restrict__ ei,
                           const float* __restrict__ asrc,
                           const float* __restrict__ adst,
                           const unsigned* __restrict__ menc,
                           const float* __restrict__ denom,
                           const float* __restrict__ h,
                           float* __restrict__ gacc) {
  long long wid = ((long long)blockIdx.x * blockDim.x + threadIdx.x) >> 5;
  int lane = threadIdx.x & 31;
  long long nwaves = ((long long)gridDim.x * blockDim.x) >> 5;
  for (long long e = wid; e < NEc + NXc; e += nwaves) {
    int s, d; edge_pair(ei, e, s, d);
    float l = edge_logit(asrc, adst, s, d);
    float alpha = __expf(l - decf(menc[d])) / denom[d];
    const float* hs = h + (size_t)s * NFP;
    float* gd = gacc + (size_t)d * NFP;
    if (lane < NFc) atomicAdd(&gd[lane], alpha * hs[lane]);
    int f2 = lane + 32;
    if (f2 < NFc) atomicAdd(&gd[f2], alpha * hs[f2]);
  }
}

// ------------------ xf = softmax(gat_out + bias), padded --------------------
__global__ void k_xf(const float* __restrict__ gacc, const float* __restrict__ bias,
                     float* __restrict__ xf) {
  int n = blockIdx.x * blockDim.x + threadIdx.x;
  if (n >= NXc) return;
  float v[NFc];
  float mx = -3.0e38f;
#pragma unroll
  for (int f = 0; f < NFc; ++f) {
    v[f] = gacc[(size_t)n * NFP + f] + bias[f];
    mx = fmaxf(mx, v[f]);
  }
  float sum = 0.0f;
#pragma unroll
  for (int f = 0; f < NFc; ++f) { v[f] = __expf(v[f] - mx); sum += v[f]; }
  float inv = 1.0f / sum;
#pragma unroll
  for (int f = 0; f < NFc; ++f) xf[(size_t)n * NFP + f] = v[f] * inv;
#pragma unroll
  for (int f = NFc; f < NFP; ++f) xf[(size_t)n * NFP + f] = 0.0f;
}

// --------------- vis_pre += gamma^T @ xf  (f32 WMMA, split-K) ---------------
// One wave owns 16 V-rows x all 64 F-cols (4 accumulators) for one K-chunk:
// gamma is read exactly once (800 MB total), 12500 waves keep HBM saturated,
// and A/B fragments are double-buffered so loads overlap the WMMAs.
__global__ void k_gemm_gammaT(const float* __restrict__ gamma,
                              const float* __restrict__ xf,
                              float* __restrict__ vis) {
  const int MT = NVc / 16;                       // 250 V-tiles
  int wave = (int)((blockIdx.x * blockDim.x + threadIdx.x) >> 5);
  int lane = threadIdx.x & 31;
  if (wave >= MT * KSPLIT) return;               // wave-uniform
  int m0    = (wave % MT) * 16;
  int chunk = wave / MT;
  const int rows = NXc / KSPLIT;                 // 1000
  int kbeg = chunk * rows;
  int kend = kbeg + rows;
  int l15 = lane & 15;
  int kh  = (lane >> 4) * 2;                     // K-pair select (ISA A/B layout)

  v8f c[4] = {v8f{}, v8f{}, v8f{}, v8f{}};

  auto loadAB = [&](int x0, v2f& a, v2f* b) {
    a.x = gamma[(size_t)(x0 + kh)     * NVc + m0 + l15];   // A[m,k] = gamma[x,v]
    a.y = gamma[(size_t)(x0 + kh + 1) * NVc + m0 + l15];
#pragma unroll
    for (int t = 0; t < 4; ++t) {
      b[t].x = xf[(size_t)(x0 + kh)     * NFP + t * 16 + l15];
      b[t].y = xf[(size_t)(x0 + kh + 1) * NFP + t * 16 + l15];
    }
  };

  v2f a0, b0[4];
  loadAB(kbeg, a0, b0);
  int x0 = kbeg;
  for (; x0 + 4 < kend; x0 += 4) {
    if (x0 + 64 < NXc)
      __builtin_prefetch(gamma + (size_t)(x0 + 64) * NVc + m0 + l15, 0, 1);
    v2f a1, b1[4];
    loadAB(x0 + 4, a1, b1);                      // next fragments in flight
#pragma unroll
    for (int t = 0; t < 4; ++t)
      c[t] = __builtin_amdgcn_wmma_f32_16x16x4_f32(false, a0, false, b0[t],
                                                   (short)0, c[t], false, false);
    a0 = a1;
#pragma unroll
    for (int t = 0; t < 4; ++t) b0[t] = b1[t];
  }
#pragma unroll
  for (int t = 0; t < 4; ++t)                    // peeled last k-step
    c[t] = __builtin_amdgcn_wmma_f32_16x16x4_f32(false, a0, false, b0[t],
                                                 (short)0, c[t], false, false);

#pragma unroll
  for (int t = 0; t < 4; ++t)
#pragma unroll
    for (int i = 0; i < 8; ++i) {
      int row = m0 + i + 8 * (lane >> 4);
      atomicAdd(&vis[(size_t)row * NFP + t * 16 + l15], c[t][i]);
    }
}

// -------------------- vis = row softmax (in place, padded) ------------------
__global__ void k_vis_softmax(float* __restrict__ vis) {
  int n = blockIdx.x * blockDim.x + threadIdx.x;
  if (n >= NVc) return;
  float v[NFc];
  float mx = -3.0e38f;
#pragma unroll
  for (int f = 0; f < NFc; ++f) { v[f] = vis[(size_t)n * NFP + f]; mx = fmaxf(mx, v[f]); }
  float sum = 0.0f;
#pragma unroll
  for (int f = 0; f < NFc; ++f) { v[f] = __expf(v[f] - mx); sum += v[f]; }
  float inv = 1.0f / sum;
#pragma unroll
  for (int f = 0; f < NFc; ++f) vis[(size_t)n * NFP + f] = v[f] * inv;
#pragma unroll
  for (int f = NFc; f < NFP; ++f) vis[(size_t)n * NFP + f] = 0.0f;
}

// ---------- w_full = softmax(W, axis=1) * exp(gene_sf), K padded ------------
__global__ void k_wfull(const float* __restrict__ W, const float* __restrict__ gsf,
                        float* __restrict__ wfull) {
  __shared__ float red[256];
  int r = blockIdx.x;           // 0..NFP-1
  int t = threadIdx.x;
  if (r >= NFc) {               // zero-pad rows 50..63 (whole block same path)
    for (int g = t; g < NGc; g += 256) wfull[(size_t)r * NGc + g] = 0.0f;
    return;
  }
  const float* row = W + (size_t)r * NGc;
  float mx = -3.0e38f;
  for (int g = t; g < NGc; g += 256) mx = fmaxf(mx, row[g]);
  red[t] = mx; __syncthreads();
  for (int off = 128; off; off >>= 1) {
    if (t < off) red[t] = fmaxf(red[t], red[t + off]);
    __syncthreads();
  }
  mx = red[0]; __syncthreads();
  float sum = 0.0f;
  for (int g = t; g < NGc; g += 256) sum += __expf(row[g] - mx);
  red[t] = sum; __syncthreads();
  for (int off = 128; off; off >>= 1) {
    if (t < off) red[t] += red[t + off];
    __syncthreads();
  }
  float inv = 1.0f / red[0];
  for (int g = t; g < NGc; g += 256)
    wfull[(size_t)r * NGc + g] = __expf(row[g] - mx) * inv * __expf(gsf[g]);
}

// ------------- xe_est = xf @ wfull[:, :NGX]  (f32 WMMA, K = 64) -------------
__global__ void k_gemm_xe(const float* __restrict__ xf, const float* __restrict__ wfull,
                          float* __restrict__ out) {
  const int NT = (NGXc + 15) / 16;      // 19 tiles (304 padded cols)
  const int MT = NXc / 16;              // 3125
  int wave = (int)((blockIdx.x * blockDim.x + threadIdx.x) >> 5);
  int lane = threadIdx.x & 31;
  if (wave >= MT * NT) return;
  int m0 = (wave / NT) * 16;
  int g0 = (wave % NT) * 16;
  int l15 = lane & 15;
  int kh  = (lane >> 4) * 2;
  v8f c = {};
#pragma unroll
  for (int k0 = 0; k0 < NFP; k0 += 4) {
    v2f a, b;
    a.x = xf[(size_t)(m0 + l15) * NFP + k0 + kh];
    a.y = xf[(size_t)(m0 + l15) * NFP + k0 + kh + 1];
    b.x = wfull[(size_t)(k0 + kh)     * NGc + g0 + l15];
    b.y = wfull[(size_t)(k0 + kh + 1) * NGc + g0 + l15];
    c = __builtin_amdgcn_wmma_f32_16x16x4_f32(false, a, false, b,
                                              (short)0, c, false, false);
  }
  int col = g0 + l15;
  if (col < NGXc) {
#pragma unroll
    for (int i = 0; i < 8; ++i) {
      int row = m0 + i + 8 * (lane >> 4);
      out[(size_t)row * NGXc + col] = c[i];
    }
  }
}

// --------- vi_est = exp(sf) * (vis @ wfull)  (f32 WMMA, K = 64) -------------
__global__ void k_gemm_vi(const float* __restrict__ vis, const float* __restrict__ wfull,
                          const float* __restrict__ sfv, float* __restrict__ out) {
  const int NT = NGc / 16;              // 1125
  const int MT = NVc / 16;              // 250
  int wave = (int)((blockIdx.x * blockDim.x + threadIdx.x) >> 5);
  int lane = threadIdx.x & 31;
  if (wave >= MT * NT) return;
  int m0 = (wave / NT) * 16;
  int g0 = (wave % NT) * 16;
  int l15 = lane & 15;
  int kh  = (lane >> 4) * 2;
  v8f c = {};
#pragma unroll
  for (int k0 = 0; k0 < NFP; k0 += 4) {
    v2f a, b;
    a.x = vis[(size_t)(m0 + l15) * NFP + k0 + kh];
    a.y = vis[(size_t)(m0 + l15) * NFP + k0 + kh + 1];
    b.x = wfull[(size_t)(k0 + kh)     * NGc + g0 + l15];
    b.y = wfull[(size_t)(k0 + kh + 1) * NGc + g0 + l15];
    c = __builtin_amdgcn_wmma_f32_16x16x4_f32(false, a, false, b,
                                              (short)0, c, false, false);
  }
#pragma unroll
  for (int i = 0; i < 8; ++i) {
    int row = m0 + i + 8 * (lane >> 4);
    out[(size_t)row * NGc + g0 + l15] = __expf(sfv[row]) * c[i];
  }
}

// ------------------------------- launcher -----------------------------------
extern "C" void kernel_launch(void* const* d_in, const int* in_sizes, int n_in,
                              void* d_out, int out_size, void* d_ws, size_t ws_size,
                              hipStream_t stream) {
  const float*     Px   = (const float*)d_in[0];
  const float*     W    = (const float*)d_in[1];
  const float*     gsf  = (const float*)d_in[2];   // gene_sf [1,NG]
  const float*     sfv  = (const float*)d_in[3];   // scale_factor_visium [NV,1]
  const float*     gamma= (const float*)d_in[4];   // [NX,NV]
  const float*     linW = (const float*)d_in[5];   // [NF,NF]
  const float*     attS = (const float*)d_in[6];
  const float*     attD = (const float*)d_in[7];
  const float*     bias = (const float*)d_in[8];
  const long long* ei   = (const long long*)d_in[9];  // int64 [2,NE]
  float* out = (float*)d_out;

  // workspace carve-out (~32 MB total)
  char* ws = (char*)d_ws;
  size_t o = 0;
  auto carve = [&](size_t bytes) -> void* {
    void* p = ws + o;
    o += (bytes + 255) & ~(size_t)255;
    return p;
  };
  float*    h     = (float*)   carve((size_t)NXc * NFP * 4);  // node features (reused as xf)
  float*    asrc  = (float*)   carve((size_t)NXc * 4);
  float*    adst  = (float*)   carve((size_t)NXc * 4);
  unsigned* menc  = (unsigned*)carve((size_t)NXc * 4);
  float*    denom = (float*)   carve((size_t)NXc * 4);
  float*    gacc  = (float*)   carve((size_t)NXc * NFP * 4);
  float*    vis   = (float*)   carve((size_t)NVc * NFP * 4);
  float*    wfull = (float*)   carve((size_t)NFP * NGc * 4);
  float*    xf    = h;  // h dead after k_edge_acc; reuse for padded xf

  k_init<<<2048, 256, 0, stream>>>(gacc, denom, menc, vis);
  k_node<<<(NXc + 255) / 256, 256, 0, stream>>>(Px, linW, attS, attD, h, asrc, adst);
  k_edge_max<<<4096, 256, 0, stream>>>(ei, asrc, adst, menc);
  k_edge_den<<<4096, 256, 0, stream>>>(ei, asrc, adst, menc, denom);
  k_edge_acc<<<4096, 256, 0, stream>>>(ei, asrc, adst, menc, denom, h, gacc);
  k_xf<<<(NXc + 255) / 256, 256, 0, stream>>>(gacc, bias, xf);

  {
    int tiles = (NVc / 16) * KSPLIT;               // 12500 waves
    k_gemm_gammaT<<<(tiles + 3) / 4, 128, 0, stream>>>(gamma, xf, vis);
  }
  k_vis_softmax<<<(NVc + 255) / 256, 256, 0, stream>>>(vis);
  k_wfull<<<NFP, 256, 0, stream>>>(W, gsf, wfull);
  {
    int tiles = (NXc / 16) * ((NGXc + 15) / 16);   // 59375
    k_gemm_xe<<<(tiles + 3) / 4, 128, 0, stream>>>(xf, wfull, out);
  }
  {
    int tiles = (NVc / 16) * (NGc / 16);           // 281250
    k_gemm_vi<<<(tiles + 3) / 4, 128, 0, stream>>>(vis, wfull, sfv,
                                                   out + (size_t)NXc * NGXc);
  }
}